// HeteroGNN_78950088835206
// MI455X (gfx1250) — compile-verified
//
#include <hip/hip_runtime.h>

typedef __attribute__((ext_vector_type(2))) float v2f;
typedef __attribute__((ext_vector_type(8))) float v8f;

#define N_USER 100000
#define N_SONG 50000
#define N_EDGE 600000
#define D_IN 128
#define H1 128
#define H2 64

#define LDS_STRIDE 260  // padded row stride (floats): conflict-free A loads, 16B-aligned rows
#define MT 32           // rows (M) per block = 2 WMMA M-tiles

// ---------------------------------------------------------------------------
// Edge aggregation kernels
// ---------------------------------------------------------------------------
__global__ __launch_bounds__(256) void count_kernel(const int* __restrict__ dst,
                                                    float* __restrict__ cnt, int E) {
  int e = blockIdx.x * blockDim.x + threadIdx.x;
  if (e < E) atomicAdd(&cnt[dst[e]], 1.0f);
}

// one thread per (edge, 4-col group): b128 load + 4 float atomics
__global__ __launch_bounds__(256) void scatter_kernel(const float* __restrict__ xsrc,
                                                      const int* __restrict__ src,
                                                      const int* __restrict__ dst,
                                                      float* __restrict__ sums, int E) {
  long long idx = (long long)blockIdx.x * blockDim.x + threadIdx.x;
  long long total = (long long)E * 32;
  if (idx >= total) return;
  int e = (int)(idx >> 5);
  int g = (int)(idx & 31);
  const float4 s = *(const float4*)(xsrc + (size_t)src[e] * D_IN + g * 4);
  float* d = sums + (size_t)dst[e] * D_IN + g * 4;
  atomicAdd(&d[0], s.x);
  atomicAdd(&d[1], s.y);
  atomicAdd(&d[2], s.z);
  atomicAdd(&d[3], s.w);
}

__global__ __launch_bounds__(256) void mean_kernel(float* __restrict__ sums,
                                                   const float* __restrict__ cnt,
                                                   long long total) {
  long long i = (long long)blockIdx.x * blockDim.x + threadIdx.x;
  if (i < total) sums[i] /= fmaxf(cnt[i >> 7], 1.0f);  // i>>7 == i / 128
}

// ---------------------------------------------------------------------------
// WMMA GEMM: out[row, 0:NOUT] = Xself[row,:]@Ws + Xmean[row,:]@Wn + bias
// K = 128 + 128 (self || mean). Block = 32 rows; each wave owns one 16-wide
// N tile and two M tiles (B fragment reused across both accumulators).
// blockDim.x = (NOUT/16)*32. Grid.x = ceil(nrows/32).
// ---------------------------------------------------------------------------
template <int NOUT, bool RELU>
__global__ __launch_bounds__(256) void sage_gemm_kernel(
    const float* __restrict__ Xself, const float* __restrict__ Xmean,
    const float* __restrict__ Ws, const float* __restrict__ Wn,
    const float* __restrict__ bias, float* __restrict__ out, int nrows) {
  __shared__ float lds[MT * LDS_STRIDE];

  const int rowBase = blockIdx.x * MT;
  const int tid = threadIdx.x;
  const int nThreads = NOUT / 16 * 32;

  // Stage A tile (32 rows x 256 cols) into LDS with float4 vectors.
  // cols [0,128) = Xself, [128,256) = Xmean
  for (int g = tid; g < MT * 256 / 4; g += nThreads) {
    int m = g >> 6;            // 64 float4-groups per 256-col row
    int kg = (g & 63) * 4;
    int row = rowBase + m;
    float4 v = make_float4(0.f, 0.f, 0.f, 0.f);
    if (row < nrows) {
      v = (kg < D_IN) ? *(const float4*)(Xself + (size_t)row * D_IN + kg)
                      : *(const float4*)(Xmean + (size_t)row * D_IN + (kg - D_IN));
    }
    *(float4*)(&lds[m * LDS_STRIDE + kg]) = v;
  }
  __syncthreads();

  const int lane = tid & 31;
  const int wave = tid >> 5;     // N-tile index
  const int m16  = lane & 15;
  const int kq   = lane >> 4;    // half-wave selects K pair (ISA f32 A/B layout)
  const int ncol = wave * 16 + m16;

  // per-lane bases; all subsequent offsets are compile-time immediates
  const float* la0 = &lds[m16 * LDS_STRIDE + 2 * kq];         // M tile 0
  const float* la1 = &lds[(m16 + 16) * LDS_STRIDE + 2 * kq];  // M tile 1
  const float* Bs = Ws + (size_t)(2 * kq) * NOUT + ncol;
  const float* Bn = Wn + (size_t)(2 * kq) * NOUT + ncol;

  v8f acc0 = {};
  v8f acc1 = {};

  // self-features x W_self
#pragma unroll
  for (int k0 = 0; k0 < D_IN; k0 += 4) {
    v2f b;
    b.x = Bs[k0 * NOUT];
    b.y = Bs[(k0 + 1) * NOUT];
    v2f a0;
    a0.x = la0[k0];
    a0.y = la0[k0 + 1];
    v2f a1;
    a1.x = la1[k0];
    a1.y = la1[k0 + 1];
    acc0 = __builtin_amdgcn_wmma_f32_16x16x4_f32(false, a0, false, b, (short)0,
                                                 acc0, false, false);
    acc1 = __builtin_amdgcn_wmma_f32_16x16x4_f32(false, a1, false, b, (short)0,
                                                 acc1, false, false);
  }
  // mean-features x W_neigh (LDS cols 128..255)
#pragma unroll
  for (int k0 = 0; k0 < D_IN; k0 += 4) {
    v2f b;
    b.x = Bn[k0 * NOUT];
    b.y = Bn[(k0 + 1) * NOUT];
    v2f a0;
    a0.x = la0[D_IN + k0];
    a0.y = la0[D_IN + k0 + 1];
    v2f a1;
    a1.x = la1[D_IN + k0];
    a1.y = la1[D_IN + k0 + 1];
    acc0 = __builtin_amdgcn_wmma_f32_16x16x4_f32(false, a0, false, b, (short)0,
                                                 acc0, false, false);
    acc1 = __builtin_amdgcn_wmma_f32_16x16x4_f32(false, a1, false, b, (short)0,
                                                 acc1, false, false);
  }

  // D layout (f32 16x16): VGPR r -> row (r + 8*halfwave), col = lane&15 + tile*16
  const float bv = bias[ncol];
#pragma unroll
  for (int r = 0; r < 8; ++r) {
    int row0 = rowBase + r + 8 * kq;
    if (row0 < nrows) {
      float v = acc0[r] + bv;
      if (RELU) v = fmaxf(v, 0.0f);
      out[(size_t)row0 * NOUT + ncol] = v;
    }
    int row1 = rowBase + 16 + r + 8 * kq;
    if (row1 < nrows) {
      float v = acc1[r] + bv;
      if (RELU) v = fmaxf(v, 0.0f);
      out[(size_t)row1 * NOUT + ncol] = v;
    }
  }
}

// ---------------------------------------------------------------------------
extern "C" void kernel_launch(void* const* d_in, const int* in_sizes, int n_in,
                              void* d_out, int out_size, void* d_ws, size_t ws_size,
                              hipStream_t stream) {
  const float* x_user = (const float*)d_in[0];
  const float* x_song = (const float*)d_in[1];
  const int* us_src = (const int*)d_in[2];
  const int* us_dst = (const int*)d_in[3];
  const int* su_src = (const int*)d_in[4];
  const int* su_dst = (const int*)d_in[5];
  const float* W1_us_neigh = (const float*)d_in[6];
  const float* W1_us_self  = (const float*)d_in[7];
  const float* b1_us       = (const float*)d_in[8];
  const float* W1_su_neigh = (const float*)d_in[9];
  const float* W1_su_self  = (const float*)d_in[10];
  const float* b1_su       = (const float*)d_in[11];
  const float* W2_us_neigh = (const float*)d_in[12];
  const float* W2_us_self  = (const float*)d_in[13];
  const float* b2_us       = (const float*)d_in[14];
  const float* W2_su_neigh = (const float*)d_in[15];
  const float* W2_su_self  = (const float*)d_in[16];
  const float* b2_su       = (const float*)d_in[17];
  float* out = (float*)d_out;

  // carve workspace
  char* ws = (char*)d_ws;
  size_t off = 0;
  auto carve = [&](size_t bytes) -> void* {
    void* p = ws + off;
    off = (off + bytes + 255) & ~(size_t)255;
    return p;
  };
  float* agg_song = (float*)carve((size_t)N_SONG * D_IN * 4);  // reused both layers
  float* agg_user = (float*)carve((size_t)N_USER * D_IN * 4);
  float* h_song   = (float*)carve((size_t)N_SONG * H1 * 4);
  float* h_user   = (float*)carve((size_t)N_USER * H1 * 4);
  float* cnt_song = (float*)carve((size_t)N_SONG * 4);
  float* cnt_user = (float*)carve((size_t)N_USER * 4);

  const int E = N_EDGE;
  const long long scatterThreads = (long long)E * 32;
  const int scatterBlocks = (int)((scatterThreads + 255) / 256);
  const int edgeBlocks = (E + 255) / 256;

  // ---- degree counts (same both layers) ----
  hipMemsetAsync(cnt_song, 0, (size_t)N_SONG * 4, stream);
  hipMemsetAsync(cnt_user, 0, (size_t)N_USER * 4, stream);
  count_kernel<<<edgeBlocks, 256, 0, stream>>>(us_dst, cnt_song, E);
  count_kernel<<<edgeBlocks, 256, 0, stream>>>(su_dst, cnt_user, E);

  // ---- layer 1 aggregation ----
  hipMemsetAsync(agg_song, 0, (size_t)N_SONG * D_IN * 4, stream);
  hipMemsetAsync(agg_user, 0, (size_t)N_USER * D_IN * 4, stream);
  scatter_kernel<<<scatterBlocks, 256, 0, stream>>>(x_user, us_src, us_dst, agg_song, E);
  scatter_kernel<<<scatterBlocks, 256, 0, stream>>>(x_song, su_src, su_dst, agg_user, E);
  mean_kernel<<<(int)(((long long)N_SONG * D_IN + 255) / 256), 256, 0, stream>>>(
      agg_song, cnt_song, (long long)N_SONG * D_IN);
  mean_kernel<<<(int)(((long long)N_USER * D_IN + 255) / 256), 256, 0, stream>>>(
      agg_user, cnt_user, (long long)N_USER * D_IN);

  // ---- layer 1 GEMMs (+ReLU), H1 = 128 -> 8 waves, 32 rows/block ----
  sage_gemm_kernel<H1, true><<<(N_SONG + MT - 1) / MT, 256, 0, stream>>>(
      x_song, agg_song, W1_us_self, W1_us_neigh, b1_us, h_song, N_SONG);
  sage_gemm_kernel<H1, true><<<(N_USER + MT - 1) / MT, 256, 0, stream>>>(
      x_user, agg_user, W1_su_self, W1_su_neigh, b1_su, h_user, N_USER);

  // ---- layer 2 aggregation (reuse agg buffers, counts unchanged) ----
  hipMemsetAsync(agg_song, 0, (size_t)N_SONG * D_IN * 4, stream);
  hipMemsetAsync(agg_user, 0, (size_t)N_USER * D_IN * 4, stream);
  scatter_kernel<<<scatterBlocks, 256, 0, stream>>>(h_user, us_src, us_dst, agg_song, E);
  scatter_kernel<<<scatterBlocks, 256, 0, stream>>>(h_song, su_src, su_dst, agg_user, E);
  mean_kernel<<<(int)(((long long)N_SONG * D_IN + 255) / 256), 256, 0, stream>>>(
      agg_song, cnt_song, (long long)N_SONG * D_IN);
  mean_kernel<<<(int)(((long long)N_USER * D_IN + 255) / 256), 256, 0, stream>>>(
      agg_user, cnt_user, (long long)N_USER * D_IN);

  // ---- layer 2 GEMMs, H2 = 64 -> 4 waves; write concat([user, song]) ----
  sage_gemm_kernel<H2, false><<<(N_USER + MT - 1) / MT, 128, 0, stream>>>(
      h_user, agg_user, W2_su_self, W2_su_neigh, b2_su, out, N_USER);
  sage_gemm_kernel<H2, false><<<(N_SONG + MT - 1) / MT, 128, 0, stream>>>(
      h_song, agg_song, W2_us_self, W2_us_neigh, b2_us, out + (size_t)N_USER * H2,
      N_SONG);
}